// CPNet_1563368096435
// MI455X (gfx1250) — compile-verified
//
#include <hip/hip_runtime.h>
#include <math.h>

typedef float v2f __attribute__((ext_vector_type(2)));
typedef float v8f __attribute__((ext_vector_type(8)));

#define NPTS 8192
#define TILE 16
#define WPB  8          // waves per block (wave32)
#define EPSV   1e-05f
#define FACTV  2.0f
#define PDEPS  1e-06f

// One wave handles a 16-row tile of pc1, streaming all of pc2 (in LDS) in
// 16-column WMMA tiles with an online (flash-style) softmax. wave32 only.
__global__ __launch_bounds__(256) void cpnet_soft_knn(
    const float* __restrict__ pc1, const float* __restrict__ pc2,
    float* __restrict__ out, float* __restrict__ partial)
{
    __shared__ __align__(16) float l2[3 * NPTS];   // 96 KB: whole pc2

    const int tid = threadIdx.x;
    // cooperative stage: global -> LDS (pc2 is [3][NPTS] contiguous)
    {
        const float4* src = (const float4*)pc2;
        float4*       dst = (float4*)l2;
        #pragma unroll 4
        for (int i = tid; i < (3 * NPTS) / 4; i += 256) dst[i] = src[i];
    }
    __syncthreads();

    const int lane = tid & 31;
    const int wave = tid >> 5;
    const int half = lane >> 4;     // 0: lanes 0-15, 1: lanes 16-31
    const int l16  = lane & 15;
    const int tile = blockIdx.x * WPB + wave;
    const int nrow = tile * TILE + l16;          // pc1 point owned by this lane

    const float x1 = pc1[nrow];
    const float y1 = pc1[NPTS + nrow];
    const float z1 = pc1[2 * NPTS + nrow];
    const float sq1 = x1 * x1 + y1 * y1 + z1 * z1;

    // B operand (pc1 tile, 4x16): lanes 0-15 hold K=0,1 ; lanes 16-31 hold K=2,3
    v2f b;
    b[0] = half ? z1   : x1;
    b[1] = half ? 0.0f : y1;

    // online softmax state for point nrow (each half covers disjoint j's)
    float m = -3.4e38f, s = 0.0f, ax = 0.0f, ay = 0.0f, az = 0.0f;

    for (int jt = 0; jt < NPTS / TILE; ++jt) {
        // A operand (pc2 tile, 16x4): lane holds row M=l16 of the tile
        const int ja = jt * TILE + l16;
        const float xa = l2[ja], ya = l2[NPTS + ja], za = l2[2 * NPTS + ja];
        v2f a;
        a[0] = half ? za   : xa;
        a[1] = half ? 0.0f : ya;

        v8f c = {};
        // D[m][n] = pc2_m . pc1_n  (m = v + 8*half within the j-tile)
        v8f d = __builtin_amdgcn_wmma_f32_16x16x4_f32(
            false, a, false, b, (short)0, c, false, false);

        // the 8 pc2 points this lane accumulates: j = jt*16 + 8*half + v
        const int jb = jt * TILE + half * 8;
        const float4 X0 = *(const float4*)&l2[jb];
        const float4 X1 = *(const float4*)&l2[jb + 4];
        const float4 Y0 = *(const float4*)&l2[NPTS + jb];
        const float4 Y1 = *(const float4*)&l2[NPTS + jb + 4];
        const float4 Z0 = *(const float4*)&l2[2 * NPTS + jb];
        const float4 Z1 = *(const float4*)&l2[2 * NPTS + jb + 4];
        const float xs[8] = {X0.x, X0.y, X0.z, X0.w, X1.x, X1.y, X1.z, X1.w};
        const float ys[8] = {Y0.x, Y0.y, Y0.z, Y0.w, Y1.x, Y1.y, Y1.z, Y1.w};
        const float zs[8] = {Z0.x, Z0.y, Z0.z, Z0.w, Z1.x, Z1.y, Z1.z, Z1.w};

        float f[8];
        float tmax = -3.4e38f;
        #pragma unroll
        for (int v = 0; v < 8; ++v) {
            const float sq2 = xs[v] * xs[v] + ys[v] * ys[v] + zs[v] * zs[v];
            float d2 = sq1 + sq2 - 2.0f * d[v];
            d2 = fmaxf(d2, EPSV);
            // FACT * 1/clamp(d2): single v_rcp_f32 (1 ulp) instead of the
            // 8-op IEEE divide sequence -- this feeds exp, so 1 ulp is plenty.
            f[v] = FACTV * __builtin_amdgcn_rcpf(d2);
            tmax = fmaxf(tmax, f[v]);
        }
        const float nm = fmaxf(m, tmax);
        const float sc = __expf(m - nm);          // 0 on first tile
        s *= sc; ax *= sc; ay *= sc; az *= sc;
        #pragma unroll
        for (int v = 0; v < 8; ++v) {
            const float w = __expf(f[v] - nm);
            s += w;
            ax = fmaf(w, xs[v], ax);
            ay = fmaf(w, ys[v], ay);
            az = fmaf(w, zs[v], az);
        }
        m = nm;
    }

    // merge the two wave-halves (same pc1 point, disjoint j sets)
    {
        const float mo  = __shfl_xor(m,  16, 32);
        const float so  = __shfl_xor(s,  16, 32);
        const float axo = __shfl_xor(ax, 16, 32);
        const float ayo = __shfl_xor(ay, 16, 32);
        const float azo = __shfl_xor(az, 16, 32);
        const float nm = fmaxf(m, mo);
        const float sA = __expf(m - nm), sB = __expf(mo - nm);
        s  = s  * sA + so  * sB;
        ax = ax * sA + axo * sB;
        ay = ay * sA + ayo * sB;
        az = az * sA + azo * sB;
    }

    const float inv = 1.0f / s;      // exact divide: once per point
    const float px = ax * inv, py = ay * inv, pz = az * inv;
    if (half == 0) {
        out[nrow]            = px;
        out[NPTS + nrow]     = py;
        out[2 * NPTS + nrow] = pz;
    }

    // per-point residual norm, deterministic per-tile partial sum
    const float dx = x1 - px + PDEPS;
    const float dy = y1 - py + PDEPS;
    const float dz = z1 - pz + PDEPS;
    float p = (half == 0) ? sqrtf(dx * dx + dy * dy + dz * dz) : 0.0f;
    p += __shfl_xor(p, 1, 32);
    p += __shfl_xor(p, 2, 32);
    p += __shfl_xor(p, 4, 32);
    p += __shfl_xor(p, 8, 32);
    p += __shfl_xor(p, 16, 32);
    if (lane == 0) partial[tile] = p;
}

__global__ __launch_bounds__(256) void cpnet_mean_reduce(
    const float* __restrict__ partial, float* __restrict__ out,
    int ntiles, float invN)
{
    __shared__ float red[256];
    float acc = 0.0f;
    for (int i = threadIdx.x; i < ntiles; i += 256) acc += partial[i];
    red[threadIdx.x] = acc;
    __syncthreads();
    for (int off = 128; off > 0; off >>= 1) {
        if ((int)threadIdx.x < off) red[threadIdx.x] += red[threadIdx.x + off];
        __syncthreads();
    }
    if (threadIdx.x == 0) out[3 * NPTS] = red[0] * invN;
}

extern "C" void kernel_launch(void* const* d_in, const int* in_sizes, int n_in,
                              void* d_out, int out_size, void* d_ws, size_t ws_size,
                              hipStream_t stream) {
    (void)n_in; (void)out_size; (void)ws_size;
    const float* pc1 = (const float*)d_in[0];   // [3, 8192] f32
    const float* pc2 = (const float*)d_in[1];   // [3, 8192] f32
    float* out = (float*)d_out;                 // 24576 (pc_nearest) + 1 (mean)
    float* ws  = (float*)d_ws;                  // >= 512 floats of scratch

    const int n      = in_sizes[0] / 3;         // 8192
    const int tiles  = n / TILE;                // 512 waves
    const int blocks = tiles / WPB;             // 64 blocks x 256 threads

    cpnet_soft_knn<<<blocks, 256, 0, stream>>>(pc1, pc2, out, ws);
    cpnet_mean_reduce<<<1, 256, 0, stream>>>(ws, out, tiles, 1.0f / (float)n);
}